// NeuralNearestNeighbors_29841432773339
// MI455X (gfx1250) — compile-verified
//
#include <hip/hip_runtime.h>
#include <math.h>

// NeuralNearestNeighbors continuous-relaxation sampler, CDNA5 (gfx1250).
// One wave32 per row of 512 logits; row staged to LDS via
// global_load_async_to_lds_b128 (ASYNCcnt), K=8 rounds computed in the
// linear domain (q <- q*(1-p)), output streamed with NT b128 stores.
// Cross-lane reductions use ds_swizzle_b32 xor-butterflies (all lanes live).

namespace {
constexpr int O        = 512;     // row length (o)
constexpr int KITER    = 8;       // K neighbor volumes
constexpr int WPB      = 8;       // waves per block
constexpr int PER_LANE = O / 32;  // 16 elements per lane

typedef __attribute__((ext_vector_type(4))) float v4f;

// ds_swizzle_b32 group-of-32 xor mode: offset = xor_mask<<10 | and_mask(0x1f)
template <int XOR_MASK>
__device__ __forceinline__ float swz_xor(float v) {
  int r = __builtin_amdgcn_ds_swizzle(__float_as_int(v),
                                      (XOR_MASK << 10) | 0x1f);
  return __int_as_float(r);
}

__device__ __forceinline__ float wave_max32(float v) {
  v = fmaxf(v, swz_xor<1>(v));
  v = fmaxf(v, swz_xor<2>(v));
  v = fmaxf(v, swz_xor<4>(v));
  v = fmaxf(v, swz_xor<8>(v));
  v = fmaxf(v, swz_xor<16>(v));
  return v;
}

__device__ __forceinline__ float wave_sum32(float v) {
  v += swz_xor<1>(v);
  v += swz_xor<2>(v);
  v += swz_xor<4>(v);
  v += swz_xor<8>(v);
  v += swz_xor<16>(v);
  return v;
}
}  // namespace

__global__ __launch_bounds__(WPB * 32, 1) void nnn_rows_kernel(
    const float* __restrict__ D, const float* __restrict__ log_temp,
    float* __restrict__ out, int rows) {
  __shared__ __align__(16) float smem[WPB * O];

  const int lane = threadIdx.x & 31;
  const int wave = threadIdx.x >> 5;
  const int row  = blockIdx.x * WPB + wave;
  if (row >= rows) return;  // wave-uniform

  // ---- Stage this wave's 2KB row into LDS via async LDS loads (gfx1250).
  // 4 x (32 lanes x 16B) = 2048B. Per-lane LDS offset = low 32 bits of the
  // shared pointer's flat address (workgroup-relative, as the ISA requires).
  const float*   rowp  = D + (size_t)row * O;
  const unsigned lbase = (unsigned)(size_t)(&smem[wave * O]);
#pragma unroll
  for (int c = 0; c < 4; ++c) {
    unsigned     loff = lbase + (unsigned)((c * 32 + lane) * 16);
    const float* g    = rowp + (c * 32 + lane) * 4;
    asm volatile("global_load_async_to_lds_b128 %0, %1, off"
                 :: "v"(loff), "v"(g)
                 : "memory");
  }
  asm volatile("s_wait_asynccnt 0" ::: "memory");

  const float inv_temp = __expf(-log_temp[0]);  // temperature = exp(log_temp)

  // ---- q_j = exp(l_j - rowmax); the whole K-loop then stays linear-domain:
  // log1mexp(log_softmax(l)) == log(1-p), so logits-update == q *= (1-p).
  float q[PER_LANE];
  {
    float l[PER_LANE];
    float m = -INFINITY;
#pragma unroll
    for (int j = 0; j < PER_LANE; ++j) {
      l[j] = smem[wave * O + j * 32 + lane] * inv_temp;
      m    = fmaxf(m, l[j]);
    }
    m = wave_max32(m);
#pragma unroll
    for (int j = 0; j < PER_LANE; ++j) q[j] = __expf(l[j] - m);
  }

  // ---- K=8 rounds: p = q/S ; q <- q*(1-p). Output layout (row, o, K):
  // out[(row*512 + e)*8 + k]. Batch 4 k-values per element -> b128 NT stores.
  float* orow = out + (size_t)row * O * KITER;
#pragma unroll
  for (int half = 0; half < 2; ++half) {
    float p[PER_LANE][4];
#pragma unroll
    for (int kk = 0; kk < 4; ++kk) {
      float s = 0.0f;
#pragma unroll
      for (int j = 0; j < PER_LANE; ++j) s += q[j];
      s = wave_sum32(s);
      const float invS = __builtin_amdgcn_rcpf(s);  // v_rcp_f32, ~1 ulp
#pragma unroll
      for (int j = 0; j < PER_LANE; ++j) {
        const float pj = q[j] * invS;
        p[j][kk]       = pj;
        q[j]           = fmaf(-q[j], pj, q[j]);  // q *= (1 - p), exact form
      }
    }
#pragma unroll
    for (int j = 0; j < PER_LANE; ++j) {
      v4f  v   = {p[j][0], p[j][1], p[j][2], p[j][3]};
      v4f* dst =
          (v4f*)(orow + (size_t)(j * 32 + lane) * KITER + (size_t)half * 4);
      __builtin_nontemporal_store(v, dst);  // 256MB stream > 192MB L2: NT
    }
  }
}

extern "C" void kernel_launch(void* const* d_in, const int* in_sizes, int n_in,
                              void* d_out, int out_size, void* d_ws,
                              size_t ws_size, hipStream_t stream) {
  const float* D        = (const float*)d_in[0];
  const float* log_temp = (const float*)d_in[1];
  float*       out      = (float*)d_out;

  const int rows   = in_sizes[0] / O;  // b*m = 16384 for reference shapes
  const int blocks = (rows + WPB - 1) / WPB;
  nnn_rows_kernel<<<blocks, WPB * 32, 0, stream>>>(D, log_temp, out, rows);
}